// PhyloConv1D_26594437496936
// MI455X (gfx1250) — compile-verified
//
#include <hip/hip_runtime.h>

typedef unsigned long long u64;
typedef unsigned int u32;

typedef __attribute__((ext_vector_type(2))) float v2f;
typedef __attribute__((ext_vector_type(4))) float f4;
typedef __attribute__((ext_vector_type(8))) float v8f;

#define F_DIM 4096
#define C_DIM 4
#define B_DIM 64
#define O_DIM 16
#define K_DIM 8
#define CF (C_DIM * F_DIM)
#define CK (C_DIM * K_DIM) // 32

__device__ __forceinline__ u64 shfl_xor_u64(u64 x, int mask) {
  int lo = __shfl_xor((int)(u32)x, mask, 32);
  int hi = __shfl_xor((int)(u32)(x >> 32), mask, 32);
  return ((u64)(u32)hi << 32) | (u32)lo;
}

__device__ __forceinline__ void ce_asc(u64 &a, u64 &b) {
  u64 t = a;
  if (b < t) { a = b; b = t; }
}

// ---------------------------------------------------------------------------
// Kernel 1: per (c,f) row, top-8 smallest distances -> sorted indices.
// One wave32 per row; 256 MB streamed with NT hints (won't be reused).
// Key = (order-preserving float bits << 32) | column  => ascending value,
// ties broken by lower index, matching jax.lax.top_k(-d, 8).
// ---------------------------------------------------------------------------
__global__ void __launch_bounds__(256)
topk8_kernel(const float *__restrict__ dist, int *__restrict__ idx_out) {
  int wid = (blockIdx.x * 256 + threadIdx.x) >> 5; // row id in [0, C*F)
  int lane = threadIdx.x & 31;
  const float *row = dist + (size_t)wid * F_DIM;

  u64 best[8];
#pragma unroll
  for (int i = 0; i < 8; ++i) best[i] = ~0ull;

#pragma unroll 4
  for (int it = 0; it < F_DIM / 128; ++it) {
    int j0 = (it * 32 + lane) * 4;
    f4 v = __builtin_nontemporal_load((const f4 *)(row + j0));
#pragma unroll
    for (int e = 0; e < 4; ++e) {
      u32 bits = __float_as_uint(v[e]);
      u32 m = bits ^ (((u32)((int)bits >> 31)) | 0x80000000u); // monotone map
      u64 key = ((u64)m << 32) | (u32)(j0 + e);
      if (key < best[7]) {
        best[7] = key;
#pragma unroll
        for (int p = 7; p > 0; --p) {
          if (best[p] < best[p - 1]) {
            u64 t = best[p]; best[p] = best[p - 1]; best[p - 1] = t;
          }
        }
      }
    }
  }

  // Butterfly all-reduce: merge sorted-8 lists, static-index bitonic network.
#pragma unroll
  for (int off = 16; off >= 1; off >>= 1) {
    u64 other[8];
#pragma unroll
    for (int i = 0; i < 8; ++i) other[i] = shfl_xor_u64(best[i], off);
    u64 L[8];
#pragma unroll
    for (int i = 0; i < 8; ++i) {
      u64 a = best[i], b = other[7 - i];
      L[i] = (a < b) ? a : b; // lowest 8 of 16, bitonic sequence
    }
    ce_asc(L[0], L[4]); ce_asc(L[1], L[5]); ce_asc(L[2], L[6]); ce_asc(L[3], L[7]);
    ce_asc(L[0], L[2]); ce_asc(L[1], L[3]); ce_asc(L[4], L[6]); ce_asc(L[5], L[7]);
    ce_asc(L[0], L[1]); ce_asc(L[2], L[3]); ce_asc(L[4], L[5]); ce_asc(L[6], L[7]);
#pragma unroll
    for (int i = 0; i < 8; ++i) best[i] = L[i];
  }

  if (lane == 0) {
    int4 r0 = make_int4((int)best[0], (int)best[1], (int)best[2], (int)best[3]);
    int4 r1 = make_int4((int)best[4], (int)best[5], (int)best[6], (int)best[7]);
    int4 *o = (int4 *)(idx_out + (size_t)wid * 8);
    o[0] = r0;
    o[1] = r1;
  }
}

// ---------------------------------------------------------------------------
// Kernel 2: gather + conv via V_WMMA_F32_16X16X4_F32 chains.
// One wave per f. M = 16 consecutive b values, N = 16 output channels,
// K = C*K = 32 reduced in 8 wmma steps of 4. Full fp32.
//   A layout: lanes 0-15 -> M=lane, K slots 0/1; lanes 16-31 -> K slots 2/3.
//   B layout: lanes 0-15 -> N=lane, K slots 0/1; lanes 16-31 -> K slots 2/3.
//   D layout: VGPR r -> M = r + (lane>=16 ? 8 : 0), N = lane & 15.
// ---------------------------------------------------------------------------
__global__ void __launch_bounds__(256)
gconv_wmma_kernel(const float *__restrict__ X, const float *__restrict__ Coord,
                  const float *__restrict__ W, const float *__restrict__ bias,
                  const int *__restrict__ idxbuf, float *__restrict__ out) {
  int f = (blockIdx.x * 256 + threadIdx.x) >> 5; // wave id == f in [0, F)
  int lane = threadIdx.x & 31;
  int half = lane >> 4;
  int mrow = lane & 15; // A row (b offset) and B/D column (o)

  // Per-lane gather offsets (c*F + neighbor column) and W fragments,
  // loop-invariant over {X,Coord} x 4 b-tiles.
  int off[16];
  float wb[16];
#pragma unroll
  for (int kk = 0; kk < 8; ++kk) {
#pragma unroll
    for (int e = 0; e < 2; ++e) {
      int ck = kk * 4 + half * 2 + e; // global reduction index, ck = c*8+k
      int c = ck >> 3;
      int k = ck & 7;
      int col = idxbuf[(c * F_DIM + f) * K_DIM + k];
      off[kk * 2 + e] = c * F_DIM + col;
      wb[kk * 2 + e] = W[mrow * CK + ck]; // W[o, c, k] flat = o*32 + ck
    }
  }
  float bo = bias[mrow];

#pragma unroll
  for (int t = 0; t < 2; ++t) {
    const float *data = t ? Coord : X;
    float *outp = out + (size_t)t * B_DIM * O_DIM * F_DIM;
#pragma unroll
    for (int b0 = 0; b0 < B_DIM; b0 += 16) {
      const float *dbase = data + (size_t)(b0 + mrow) * CF;
      v8f acc = {};
#pragma unroll
      for (int kk = 0; kk < 8; ++kk) {
        v2f a;
        a.x = dbase[off[kk * 2 + 0]];
        a.y = dbase[off[kk * 2 + 1]];
        v2f bf;
        bf.x = wb[kk * 2 + 0];
        bf.y = wb[kk * 2 + 1];
        acc = __builtin_amdgcn_wmma_f32_16x16x4_f32(
            /*neg_a=*/false, a, /*neg_b=*/false, bf,
            /*c_mod=*/(short)0, acc, /*reuse_a=*/false, /*reuse_b=*/false);
      }
#pragma unroll
      for (int r = 0; r < 8; ++r) {
        float v = acc[r] + bo;
        v = v > 0.0f ? v : 0.0f; // relu
        int mb = b0 + r + half * 8;
        outp[((size_t)mb * O_DIM + mrow) * F_DIM + f] = v;
      }
    }
  }
}

extern "C" void kernel_launch(void *const *d_in, const int *in_sizes, int n_in,
                              void *d_out, int out_size, void *d_ws,
                              size_t ws_size, hipStream_t stream) {
  (void)in_sizes; (void)n_in; (void)out_size; (void)ws_size;
  const float *X = (const float *)d_in[0];
  const float *Coord = (const float *)d_in[1];
  const float *dist = (const float *)d_in[2];
  const float *W = (const float *)d_in[3];
  const float *bias = (const float *)d_in[4];
  float *out = (float *)d_out;
  int *idxbuf = (int *)d_ws; // C*F*K ints = 512 KB

  // Phase 1: 16384 rows, one wave32 each -> 2048 blocks of 8 waves.
  topk8_kernel<<<(C_DIM * F_DIM) / 8, 256, 0, stream>>>(dist, idxbuf);

  // Phase 2: 4096 f-waves -> 512 blocks of 8 waves.
  gconv_wmma_kernel<<<F_DIM / 8, 256, 0, stream>>>(X, Coord, W, bias, idxbuf,
                                                   out);
}